// KalmanNet_283467842225
// MI455X (gfx1250) — compile-verified
//
#include <hip/hip_runtime.h>

// ---------------- problem constants ----------------
#define BTOT   524288
#define ROWS   256          // batch rows per block
#define THREADS 256         // 8 wave32 waves
#define NGRID  (BTOT / ROWS)

// LDS strides (floats), padded so 16 consecutive rows map to 16 distinct banks
#define AS     84           // A panel: k=0..79  (gru_in[13] | pad3 | h[64])
#define KS_IN  20           // i_n weight panel, K=16
#define KS_HN  68           // h_n weight panel, K=64
#define KS_FC  68           // fc weight panel,  K=64
#define HS     68           // h_new staging,    K=64
#define KMS    40           // K-matrix staging, 36 cols
#define MS     13           // misc: innov[6], p_pred[3], v_pred[3]

typedef float    v2f __attribute__((ext_vector_type(2)));
typedef float    v8f __attribute__((ext_vector_type(8)));
typedef unsigned int v4u __attribute__((ext_vector_type(4)));
typedef int      v4i __attribute__((ext_vector_type(4)));
typedef int      v8i __attribute__((ext_vector_type(8)));

// smem layout (float offsets)
constexpr int OFF_A   = 0;                         // ROWS x AS
constexpr int OFF_BRZ = OFF_A   + ROWS * AS;       // 128 x AS   (r,z cols, K=80 combined)
constexpr int OFF_BIN = OFF_BRZ + 128  * AS;       // 64  x KS_IN (i_n cols, K=16)
constexpr int OFF_BHN = OFF_BIN + 64   * KS_IN;    // 64  x KS_HN (h_n cols, K=64)
constexpr int OFF_BFC = OFF_BHN + 64   * KS_HN;    // 48  x KS_FC (fc cols padded, K=64)
constexpr int OFF_HN  = OFF_BFC + 48   * KS_FC;    // ROWS x HS
constexpr int OFF_K   = OFF_HN  + ROWS * HS;       // ROWS x KMS
constexpr int OFF_M   = OFF_K   + ROWS * KMS;      // ROWS x MS
constexpr int SMEM_FLOATS = OFF_M + ROWS * MS;     // = 72128 floats
constexpr int SMEM_BYTES  = SMEM_FLOATS * 4;       // = 288512 B  (< 320KB/WGP)

// Only meaningful in the device pass; host pass has no amdgcn builtins.
#if defined(__HIP_DEVICE_COMPILE__) && !__has_builtin(__builtin_amdgcn_wmma_f32_16x16x4_f32)
#error "need __builtin_amdgcn_wmma_f32_16x16x4_f32 for gfx1250"
#endif

__device__ __forceinline__ v8f wmma4(v2f a, v2f b, v8f c) {
  // D = A(16x4,f32) * B(4x16,f32) + C(16x16,f32)
  return __builtin_amdgcn_wmma_f32_16x16x4_f32(false, a, false, b, (short)0, c,
                                               false, false);
}

__device__ __forceinline__ float fsig(float x) {
  return __builtin_amdgcn_rcpf(1.f + __builtin_amdgcn_exp2f(-1.442695041f * x));
}
__device__ __forceinline__ float ftanh(float x) {
  return 1.f - 2.f * __builtin_amdgcn_rcpf(__builtin_amdgcn_exp2f(2.885390082f * x) + 1.f);
}

// -------- Tensor Data Mover: 2D tile (rows x dim0 f32) -> LDS, with LDS row
// padding applied every 64 DWORDs (pad codes per D# group1 tables).
// pad_interval code 5 -> pad fires every 2^(5+1)=64 DWORDs. pad_amt_code N -> N+1 DWORDs.
__device__ __forceinline__ void tdm_load_2d(unsigned int lds_byte, const void* gptr,
                                            unsigned int dim0, unsigned int rows_tensor,
                                            unsigned int tile_rows,
                                            unsigned int stride0,
                                            unsigned int pad_amt_code) {
  unsigned long long ga = (unsigned long long)gptr;
  v4u g0;
  g0[0] = 1u;                                        // count=1, user mode, no gather
  g0[1] = lds_byte;                                  // [63:32] lds_addr (bytes)
  g0[2] = (unsigned int)(ga & 0xffffffffull);        // [95:64] global_addr lo
  g0[3] = (unsigned int)((ga >> 32) & 0x01ffffffull) // [120:96] global_addr hi
          | (2u << 30);                              // [127:126] type=2 (image)
  v8i g1;
  g1[0] = (int)((2u << 16)                           // data_size = 4B
                | (1u << 20)                         // pad_enable
                | (5u << 22)                         // pad_interval: 64 DWORDs
                | (pad_amt_code << 25));             // pad_amount code
  g1[1] = (int)((dim0 & 0xffffu) << 16);             // [63:48] tensor_dim0 lo16
  g1[2] = (int)(((dim0 >> 16) & 0xffffu) | ((rows_tensor & 0xffffu) << 16));
  g1[3] = (int)(((rows_tensor >> 16) & 0xffffu) | ((dim0 & 0xffffu) << 16)); // tile_dim0 = dim0
  g1[4] = (int)(tile_rows & 0xffffu);                // tile_dim1 (tile_dim2 = 0)
  g1[5] = (int)stride0;                              // tensor_dim0_stride lo32
  g1[6] = 0;                                         // stride hi16 | dim1_stride lo16
  g1[7] = 0;
  v4i z{};
  v8i z8{};
  // 6-arg toolchain form: (g0, g1, g2, g3, <extra group>, cpol)
  __builtin_amdgcn_tensor_load_to_lds(g0, g1, z, z, z8, 0);
}

__global__ void __launch_bounds__(THREADS)
kalmannet_fused(const float* __restrict__ x_curr, const float* __restrict__ h_gru,
                const float* __restrict__ inputs, const float* __restrict__ ln_w,
                const float* __restrict__ ln_b,  const float* __restrict__ w_ih,
                const float* __restrict__ w_hh,  const float* __restrict__ b_ih,
                const float* __restrict__ b_hh,  const float* __restrict__ fc_w,
                const float* __restrict__ fc_b,  float* __restrict__ out) {
  extern __shared__ float sm[];
  const int tid    = threadIdx.x;
  const int block0 = blockIdx.x * ROWS;
  const int wv     = tid >> 5;

  // ---------------- Phase 0: kick off TDM DMA of the big panels ----------------
  // wv is wave-uniform -> scalar branch, so waves 1..7 never issue the TDM ops
  // (TDM ignores EXEC, so a lane-level 'if' would not be enough).
  if (wv == 0) {
    unsigned int base = (unsigned int)(unsigned long long)(&sm[0]);
    // h_gru block tile (256x64) -> A panel cols 16..79, LDS row stride 84 (pad 20 dw)
    tdm_load_2d(base + (OFF_A + 16) * 4, h_gru + (size_t)block0 * 64,
                64u, 256u, 256u, 64u, /*pad 20 dw*/ 19u);
    // w_hh rows 0..127 (r,z gates) -> BRZ cols 16..79, stride 84 (pad 20 dw)
    tdm_load_2d(base + (OFF_BRZ + 16) * 4, w_hh, 64u, 128u, 128u, 64u, 19u);
    // w_hh rows 128..191 (h_n) -> BHN, stride 68 (pad 4 dw)
    tdm_load_2d(base + OFF_BHN * 4, w_hh + 128 * 64, 64u, 64u, 64u, 64u, 3u);
    // fc_w (36x64) -> BFC as 48 rows, stride 68; rows 36..47 OOB -> zero-filled
    tdm_load_2d(base + OFF_BFC * 4, fc_w, 64u, 36u, 48u, 64u, 3u);
  }

  // ---------------- Phase 1: per-row physics + LayerNorm ----------------
  {
    const int r = tid;
    const size_t rg = (size_t)block0 + r;
    const float* xc = x_curr + rg * 10;
    const float* in = inputs + rg * 9;
    float px = xc[0], py = xc[1], pz = xc[2];
    float vx = xc[3], vy = xc[4], vz = xc[5];
    float qw = xc[6], qx = xc[7], qy = xc[8], qz = xc[9];
    float ax = in[0], ay = in[1], az = in[2];
    float gx = in[3], gy = in[4], gz = in[5];
    float mx = in[6], my = in[7], mz = in[8];

    // delta quaternion (DT*0.5 = 0.005)
    float hx = gx * 0.005f, hy = gy * 0.005f, hz = gz * 0.005f;
    float dn = __builtin_amdgcn_rsqf(1.f + hx * hx + hy * hy + hz * hz);
    float dw = dn, dx = hx * dn, dy = hy * dn, dz = hz * dn;
    // q_pred = q * dq, normalized
    float pw = qw * dw - qx * dx - qy * dy - qz * dz;
    float pxq = qw * dx + qx * dw + qy * dz - qz * dy;
    float pyq = qw * dy - qx * dz + qy * dw + qz * dx;
    float pzq = qw * dz + qx * dy - qy * dx + qz * dw;
    float pn = __builtin_amdgcn_rsqf(pw * pw + pxq * pxq + pyq * pyq + pzq * pzq);
    pw *= pn; pxq *= pn; pyq *= pn; pzq *= pn;

    // R(q) applied to accel
    float r00 = 1.f - 2.f * (qy * qy + qz * qz), r01 = 2.f * (qx * qy - qz * qw),
          r02 = 2.f * (qx * qz + qy * qw);
    float r10 = 2.f * (qx * qy + qz * qw), r11 = 1.f - 2.f * (qx * qx + qz * qz),
          r12 = 2.f * (qy * qz - qx * qw);
    float r20 = 2.f * (qx * qz - qy * qw), r21 = 2.f * (qy * qz + qx * qw),
          r22 = 1.f - 2.f * (qx * qx + qy * qy);
    float awx = r00 * ax + r01 * ay + r02 * az;
    float awy = r10 * ax + r11 * ay + r12 * az;
    float awz = r20 * ax + r21 * ay + r22 * az - 9.81f;

    float vpx = vx + awx * 0.01f, vpy = vy + awy * 0.01f, vpz = vz + awz * 0.01f;
    float ppx = px + vx * 0.01f + awx * 5e-5f;
    float ppy = py + vy * 0.01f + awy * 5e-5f;
    float ppz = pz + vz * 0.01f + awz * 5e-5f;

    // R(q_pred); inverse-rotate gravity and mag reference
    float s00 = 1.f - 2.f * (pyq * pyq + pzq * pzq), s01 = 2.f * (pxq * pyq - pzq * pw),
          s02 = 2.f * (pxq * pzq + pyq * pw);
    float s20 = 2.f * (pxq * pzq - pyq * pw), s21 = 2.f * (pyq * pzq + pxq * pw),
          s22 = 1.f - 2.f * (pxq * pxq + pyq * pyq);
    float iv0 = mx - (20.f * s00 - 40.f * s20);
    float iv1 = my - (20.f * s01 - 40.f * s21);
    float iv2 = mz - (20.f * s02 - 40.f * s22);
    float iv3 = ax - 9.81f * s20;
    float iv4 = ay - 9.81f * s21;
    float iv5 = az - 9.81f * s22;

    float gv[13] = {iv0, iv1, iv2, iv3, iv4, iv5, vpx, vpy, vpz, pw, pxq, pyq, pzq};
    float mu = 0.f;
#pragma unroll
    for (int i = 0; i < 13; ++i) mu += gv[i];
    mu *= (1.f / 13.f);
    float var = 0.f;
#pragma unroll
    for (int i = 0; i < 13; ++i) { float d = gv[i] - mu; var += d * d; }
    var *= (1.f / 13.f);
    float sc = __builtin_amdgcn_rsqf(var + 1e-5f);
#pragma unroll
    for (int i = 0; i < 13; ++i)
      sm[OFF_A + r * AS + i] = (gv[i] - mu) * sc * ln_w[i] + ln_b[i];
    sm[OFF_A + r * AS + 13] = 0.f;
    sm[OFF_A + r * AS + 14] = 0.f;
    sm[OFF_A + r * AS + 15] = 0.f;

    float* msc = &sm[OFF_M + r * MS];
    msc[0] = iv0; msc[1] = iv1; msc[2] = iv2; msc[3] = iv3; msc[4] = iv4; msc[5] = iv5;
    msc[6] = ppx; msc[7] = ppy; msc[8] = ppz; msc[9] = vpx; msc[10] = vpy; msc[11] = vpz;

    // x_new[:,6:10] = q_pred (final, no correction on quaternion)
    out[rg * 10 + 6] = pw;  out[rg * 10 + 7] = pxq;
    out[rg * 10 + 8] = pyq; out[rg * 10 + 9] = pzq;
  }

  // small irregular w_ih panels stay on the VALU path (13-wide rows + zero pad)
  for (int i = tid; i < 128 * 16; i += THREADS) {
    int n = i >> 4, k = i & 15;
    sm[OFF_BRZ + n * AS + k] = (k < 13) ? w_ih[n * 13 + k] : 0.f;
  }
  for (int i = tid; i < 64 * 16; i += THREADS) {
    int n = i >> 4, k = i & 15;
    sm[OFF_BIN + n * KS_IN + k] = (k < 13) ? w_ih[(128 + n) * 13 + k] : 0.f;
  }

  // wave0 waits for all 4 TDM transfers, then everyone syncs on the LDS panels
  if (wv == 0) __builtin_amdgcn_s_wait_tensorcnt(0);
  __syncthreads();

  // ---------------- Phase 2: WMMA GEMMs + GRU gates ----------------
  const int lane = tid & 31;
  const int lo   = lane & 15;   // row-of-A / col-of-B within tile
  const int hi   = lane >> 4;   // K sub-pair selector (ISA A/B f32 layout)

  auto ldA = [&](int mloc, int kb) -> v2f {
    return *reinterpret_cast<const v2f*>(&sm[OFF_A + (mloc + lo) * AS + kb + 2 * hi]);
  };
  auto ldB = [&](int off, int stride, int n0, int kb) -> v2f {
    return *reinterpret_cast<const v2f*>(&sm[off + (n0 + lo) * stride + kb + 2 * hi]);
  };

  for (int it = 0; it < 2; ++it) {
    const int mloc = (wv * 2 + it) * 16;   // this wave's 16-row M-tile

    for (int t = 0; t < 4; ++t) {          // 16-column gate tile: cols t*16..t*16+15
      v8f ar{}, az8{}, ain{}, ahn{};
#pragma unroll
      for (int k0 = 0; k0 < 80; k0 += 4) { // combined [gru_in|h] x [w_ih|w_hh]
        v2f a = ldA(mloc, k0);
        ar  = wmma4(a, ldB(OFF_BRZ, AS, t * 16, k0), ar);        // r-gate cols
        az8 = wmma4(a, ldB(OFF_BRZ, AS, 64 + t * 16, k0), az8);  // z-gate cols
      }
#pragma unroll
      for (int k0 = 0; k0 < 16; k0 += 4)   // i_n = gru_in @ w_ih[128:192].T
        ain = wmma4(ldA(mloc, k0), ldB(OFF_BIN, KS_IN, t * 16, k0), ain);
#pragma unroll
      for (int k0 = 0; k0 < 64; k0 += 4)   // h_n = h @ w_hh[128:192].T
        ahn = wmma4(ldA(mloc, 16 + k0), ldB(OFF_BHN, KS_HN, t * 16, k0), ahn);

      const int col = t * 16 + lo;         // 0..63
      float br  = b_ih[col] + b_hh[col];
      float bz  = b_ih[64 + col] + b_hh[64 + col];
      float bi  = b_ih[128 + col];
      float bh  = b_hh[128 + col];
#pragma unroll
      for (int j = 0; j < 8; ++j) {        // C layout: VGPR j -> row j (+8 for hi lanes)
        int rloc = mloc + j + 8 * hi;
        float rg_ = fsig(ar[j] + br);
        float zg  = fsig(az8[j] + bz);
        float ng  = ftanh(ain[j] + bi + rg_ * (ahn[j] + bh));
        float hold = sm[OFF_A + rloc * AS + 16 + col];
        float hnew = (1.f - zg) * ng + zg * hold;
        sm[OFF_HN + rloc * HS + col] = hnew;
        out[(size_t)BTOT * 10 + (size_t)(block0 + rloc) * 64 + col] = hnew;
      }
    }

    // K = h_new @ fc_w.T + fc_b   (N=36 padded to 3 tiles of 16)
    for (int nt = 0; nt < 3; ++nt) {
      v8f ak{};
#pragma unroll
      for (int k0 = 0; k0 < 64; k0 += 4) {
        v2f a = *reinterpret_cast<const v2f*>(&sm[OFF_HN + (mloc + lo) * HS + k0 + 2 * hi]);
        ak = wmma4(a, ldB(OFF_BFC, KS_FC, nt * 16, k0), ak);
      }
      const int n = nt * 16 + lo;
      float bias = (n < 36) ? fc_b[n] : 0.f;
#pragma unroll
      for (int j = 0; j < 8; ++j) {
        int rloc = mloc + j + 8 * hi;
        if (n < 36) sm[OFF_K + rloc * KMS + n] = ak[j] + bias;
      }
    }
  }
  __syncthreads();

  // ---------------- Phase 3: correction + x_new ----------------
  {
    const int r = tid;
    const size_t rg = (size_t)block0 + r;
    const float* msc = &sm[OFF_M + r * MS];
    const float* Krow = &sm[OFF_K + r * KMS];
    float iv[6];
#pragma unroll
    for (int i = 0; i < 6; ++i) iv[i] = msc[i];
#pragma unroll
    for (int i = 0; i < 6; ++i) {
      float s = 0.f;
#pragma unroll
      for (int j = 0; j < 6; ++j) s += Krow[i * 6 + j] * iv[j];
      out[rg * 10 + i] = msc[6 + i] + s;
    }
  }
}

extern "C" void kernel_launch(void* const* d_in, const int* in_sizes, int n_in,
                              void* d_out, int out_size, void* d_ws, size_t ws_size,
                              hipStream_t stream) {
  const float* x_curr = (const float*)d_in[0];
  const float* h_gru  = (const float*)d_in[1];
  const float* inputs = (const float*)d_in[2];
  const float* ln_w   = (const float*)d_in[3];
  const float* ln_b   = (const float*)d_in[4];
  const float* w_ih   = (const float*)d_in[5];
  const float* w_hh   = (const float*)d_in[6];
  const float* b_ih   = (const float*)d_in[7];
  const float* b_hh   = (const float*)d_in[8];
  const float* fc_w   = (const float*)d_in[9];
  const float* fc_b   = (const float*)d_in[10];
  float* out = (float*)d_out;

  (void)in_sizes; (void)n_in; (void)out_size; (void)d_ws; (void)ws_size;
  // CDNA5 allows up to 320KB LDS per workgroup; raise the dynamic-LDS cap.
  (void)hipFuncSetAttribute((const void*)kalmannet_fused,
                            hipFuncAttributeMaxDynamicSharedMemorySize, SMEM_BYTES);
  kalmannet_fused<<<NGRID, THREADS, SMEM_BYTES, stream>>>(
      x_curr, h_gru, inputs, ln_w, ln_b, w_ih, w_hh, b_ih, b_hh, fc_w, fc_b, out);
}